// BottleneckBlock_43679817400603
// MI455X (gfx1250) — compile-verified
//
#include <hip/hip_runtime.h>
#include <math.h>

typedef __attribute__((ext_vector_type(2))) float v2f;
typedef __attribute__((ext_vector_type(4))) float v4f;
typedef __attribute__((ext_vector_type(8))) float v8f;

static constexpr int    Nb   = 8;
static constexpr int    Wd   = 64;
static constexpr int    Tt   = 8192;
static constexpr int    KB   = 2048;
static constexpr int    ROWS = Nb * Tt;          // 65536
static constexpr int    RPW  = 32;               // rows per wave (2 x 16-row strips)
static constexpr int    STRIPS = ROWS / RPW;     // 2048 waves
static constexpr int    CTILES = KB / 16;        // 128

// workspace layout (floats)
static constexpr int    WS_COLNORM = 0;
static constexpr int    WS_SX  = KB;
static constexpr int    WS_SXX = KB + STRIPS;
static constexpr int    WS_FIT = KB + 2 * STRIPS;

// output layout (floats): x_l | x_d | commit, fit, prenorm
static constexpr size_t OUT_XL = 0;
static constexpr size_t OUT_XD = (size_t)ROWS;
static constexpr size_t OUT_SC = (size_t)ROWS + (size_t)Nb * Wd * Tt;

// ---------------- codebook norms: ||k_j||^2 ----------------
__global__ void vq_colnorm(const float* __restrict__ k, float* __restrict__ ws) {
  int j = blockIdx.x * blockDim.x + threadIdx.x;
  if (j >= KB) return;
  const v4f* kr = (const v4f*)(k + (size_t)j * Wd);  // rows are 256B aligned
  float s = 0.f;
#pragma unroll
  for (int i = 0; i < 16; ++i) {
    v4f v = kr[i];
    s += v.x * v.x + v.y * v.y + v.z * v.z + v.w * v.w;
  }
  ws[WS_COLNORM + j] = s;
}

// ---------------- main kernel: 1 wave per 32 rows (two 16-row strips) ----------------
__global__ __launch_bounds__(32) void vq_strip(const float* __restrict__ x,
                                               const float* __restrict__ k,
                                               float* __restrict__ out,
                                               float* __restrict__ ws) {
  const int lane  = threadIdx.x;        // 0..31 (wave32)
  const int strip = blockIdx.x;         // 0..2047
  const int i0    = strip * RPW;
  const int n     = i0 / Tt;
  const int t0    = i0 % Tt;            // 32-aligned; whole block shares n
  const int ml    = lane & 15;          // M (row in tile) / N (col in tile)
  const int kl2   = (lane >> 4) * 2;    // K-pair select per ISA 16x4 f32 layout

  // ---- Load two A strips (2 x 16 rows x 64 dims) -> 64 VGPRs; prenorm partials ----
  const float* xb0 = x + (size_t)n * Wd * Tt + t0 + ml;        // rows t0 .. t0+15
  const float* xb1 = xb0 + 16;                                 // rows t0+16 .. t0+31
  v2f a0[16], a1[16];
  float sx = 0.f, sxx = 0.f;
#pragma unroll
  for (int ks = 0; ks < 16; ++ks) {
    const size_t w0 = (size_t)(ks * 4 + kl2) * Tt;
    float x0 = xb0[w0], y0 = xb0[w0 + Tt];
    float x1 = xb1[w0], y1 = xb1[w0 + Tt];
    a0[ks].x = x0; a0[ks].y = y0;
    a1[ks].x = x1; a1[ks].y = y1;
    sx  += (x0 + y0) + (x1 + y1);
    sxx += (x0 * x0 + y0 * y0) + (x1 * x1 + y1 * y1);
  }

  const float* cn = ws + WS_COLNORM;
  float minv0[8], minv1[8];
  int   mini0[8], mini1[8];
#pragma unroll
  for (int r = 0; r < 8; ++r) {
    minv0[r] = 3.4e38f; mini0[r] = 0;
    minv1[r] = 3.4e38f; mini1[r] = 0;
  }

  // ---- sweep 128 column tiles: each B64 codebook load feeds TWO wmma chains ----
  for (int ct = 0; ct < CTILES; ++ct) {
    const int col = ct * 16 + ml;
    const float* kp = k + (size_t)col * Wd + kl2;
    v8f acc0 = {0.f, 0.f, 0.f, 0.f, 0.f, 0.f, 0.f, 0.f};
    v8f acc1 = {0.f, 0.f, 0.f, 0.f, 0.f, 0.f, 0.f, 0.f};
#pragma unroll
    for (int ks = 0; ks < 16; ++ks) {
      v2f b = *(const v2f*)(kp + ks * 4);   // contiguous 8B load, L2-hot codebook
      acc0 = __builtin_amdgcn_wmma_f32_16x16x4_f32(
          false, a0[ks], false, b, (short)0, acc0, false, false);
      acc1 = __builtin_amdgcn_wmma_f32_16x16x4_f32(
          false, a1[ks], false, b, (short)0, acc1, false, false);
    }
    const float c = cn[col];
#pragma unroll
    for (int r = 0; r < 8; ++r) {
      float d0 = c - 2.0f * acc0[r];        // ||k||^2 - 2 x.k (row norm added later)
      float d1 = c - 2.0f * acc1[r];
      if (d0 < minv0[r]) { minv0[r] = d0; mini0[r] = col; }  // '<' keeps first col
      if (d1 < minv1[r]) { minv1[r] = d1; mini1[r] = col; }
    }
  }

  // ---- cross-lane argmin within each 16-lane half (rows r / r+8 per strip) ----
#pragma unroll
  for (int r = 0; r < 8; ++r) {
#pragma unroll
    for (int off = 8; off >= 1; off >>= 1) {
      float ov0 = __shfl_xor(minv0[r], off, 32);
      int   oi0 = __shfl_xor(mini0[r], off, 32);
      if (ov0 < minv0[r] || (ov0 == minv0[r] && oi0 < mini0[r])) { minv0[r] = ov0; mini0[r] = oi0; }
      float ov1 = __shfl_xor(minv1[r], off, 32);
      int   oi1 = __shfl_xor(mini1[r], off, 32);
      if (ov1 < minv1[r] || (ov1 == minv1[r] && oi1 < mini1[r])) { minv1[r] = ov1; mini1[r] = oi1; }
    }
  }

  // ---- wave totals: fit-sum and prenorm partials ----
  float tmin = 0.f;
#pragma unroll
  for (int r = 0; r < 8; ++r) tmin += minv0[r] + minv1[r];
  tmin += __shfl_xor(tmin, 16, 32);            // halves hold disjoint row sets
#pragma unroll
  for (int off = 16; off >= 1; off >>= 1) {
    sx  += __shfl_xor(sx, off, 32);
    sxx += __shfl_xor(sxx, off, 32);
  }
  const float fitsum = tmin + sxx;             // sxx == sum of row norms over 32 rows

  __shared__ int s_idx[32];
  if (lane == 0) {
#pragma unroll
    for (int r = 0; r < 8; ++r) {
      s_idx[r]      = mini0[r];
      s_idx[16 + r] = mini1[r];
      out[OUT_XL + i0 + r]      = (float)mini0[r];
      out[OUT_XL + i0 + 16 + r] = (float)mini1[r];
    }
    ws[WS_SX  + strip] = sx;
    ws[WS_SXX + strip] = sxx;
    ws[WS_FIT + strip] = fitsum;
  }
  if (lane == 16) {
#pragma unroll
    for (int r = 0; r < 8; ++r) {
      s_idx[8 + r]  = mini0[r];
      s_idx[24 + r] = mini1[r];
      out[OUT_XL + i0 + 8 + r]  = (float)mini0[r];
      out[OUT_XL + i0 + 24 + r] = (float)mini1[r];
    }
  }
  __syncthreads();   // single-wave workgroup: just orders LDS

  // ---- x_d scatter: out layout (N, W, T); half-wave writes 16 consecutive t ----
#pragma unroll
  for (int g = 0; g < 2; ++g) {
    const int myidx = s_idx[g * 16 + ml];
    const float* krow = k + (size_t)myidx * Wd;
    float* ob = out + OUT_XD + (size_t)n * Wd * Tt + t0 + g * 16 + ml;
#pragma unroll
    for (int wp = 0; wp < 32; ++wp) {
      int w = wp * 2 + (lane >> 4);
      ob[(size_t)w * Tt] = krow[w];
    }
  }
}

// ---------------- deterministic fixed-order final reduction ----------------
__global__ void vq_finalize(const float* __restrict__ ws, float* __restrict__ out) {
  __shared__ float rs[256], rx[256], rxx[256];
  const int t = threadIdx.x;
  float s = 0.f, xs = 0.f, xx = 0.f;
  for (int j = t; j < STRIPS; j += 256) {   // fixed traversal order
    s  += ws[WS_FIT + j];
    xs += ws[WS_SX  + j];
    xx += ws[WS_SXX + j];
  }
  rs[t] = s; rx[t] = xs; rxx[t] = xx;
  __syncthreads();
  for (int off = 128; off > 0; off >>= 1) {
    if (t < off) { rs[t] += rs[t + off]; rx[t] += rx[t + off]; rxx[t] += rxx[t + off]; }
    __syncthreads();
  }
  if (t == 0) {
    const float sumfit = rs[0], sumx = rx[0], sumxx = rxx[0];
    const float rows = (float)ROWS;
    const float sz   = (float)((size_t)Nb * Wd * Tt);
    out[OUT_SC + 0] = sumfit / sz;                       // commit_loss = fit / W
    out[OUT_SC + 1] = sumfit / rows;                     // fit
    out[OUT_SC + 2] = sqrtf(fmaxf((sumxx - sumx * sumx / sz) / sz, 0.f)); // prenorm
  }
}

extern "C" void kernel_launch(void* const* d_in, const int* in_sizes, int n_in,
                              void* d_out, int out_size, void* d_ws, size_t ws_size,
                              hipStream_t stream) {
  const float* x = (const float*)d_in[0];   // (N, W, T) f32
  const float* k = (const float*)d_in[1];   // (K_BINS, W) f32
  float* out = (float*)d_out;
  float* ws  = (float*)d_ws;                // needs (2048 + 3*2048) floats = 32 KB

  vq_colnorm<<<KB / 256, 256, 0, stream>>>(k, ws);
  vq_strip<<<STRIPS, 32, 0, stream>>>(x, k, out, ws);
  vq_finalize<<<1, 256, 0, stream>>>(ws, out);
}